// StructuralFeatures_34505767256483
// MI455X (gfx1250) — compile-verified
//
#include <hip/hip_runtime.h>
#include <math.h>

#define B_   4
#define N_   2048
#define K_   30

typedef __attribute__((ext_vector_type(2))) float v2f;
typedef __attribute__((ext_vector_type(8))) float v8f;

// ---------- small float3 helpers ----------
__device__ __forceinline__ float3 f3(float x, float y, float z) { return make_float3(x, y, z); }
__device__ __forceinline__ float3 sub3(float3 a, float3 b) { return f3(a.x-b.x, a.y-b.y, a.z-b.z); }
__device__ __forceinline__ float3 add3(float3 a, float3 b) { return f3(a.x+b.x, a.y+b.y, a.z+b.z); }
__device__ __forceinline__ float3 scl3(float3 a, float s)  { return f3(a.x*s, a.y*s, a.z*s); }
__device__ __forceinline__ float  dot3(float3 a, float3 b) { return a.x*b.x + a.y*b.y + a.z*b.z; }
__device__ __forceinline__ float3 cross3(float3 a, float3 b) {
  return f3(a.y*b.z - a.z*b.y, a.z*b.x - a.x*b.z, a.x*b.y - a.y*b.x);
}
__device__ __forceinline__ float3 norm3(float3 a) {  // reference normalize(): eps=1e-8 inside sqrt(max(...))
  float n = sqrtf(fmaxf(dot3(a,a), 1e-8f));
  return f3(a.x/n, a.y/n, a.z/n);
}

// =====================================================================
// Kernel 1: pairwise CA distances via V_WMMA_F32_16X16X4_F32 + top-K.
// Block = 256 threads (8 waves), one block per (batch, 16-row stripe).
// Distances for the stripe live in LDS (16 x 2048 f32 = 128KB of the
// WGP's 320KB). Top-K = 30 successive (value,index)-packed u64 min
// reductions per row (matches jax.lax.top_k tie-breaking exactly).
// =====================================================================
__global__ void k_dist_topk(const float* __restrict__ X, const float* __restrict__ mask,
                            float* __restrict__ dnbr, int* __restrict__ eidx,
                            float* __restrict__ eidxf)
{
  extern __shared__ float smem[];
  float* ldsD = smem;               // [16][N_]
  float* Q    = smem + 16 * N_;     // [16][4] query coords (x,y,z,0)
  float* qn   = Q + 64;             // [16] |q|^2
  float* mq   = qn + 16;            // [16] query mask

  const int tid = threadIdx.x;
  const int b   = blockIdx.x / (N_ / 16);
  const int i0  = (blockIdx.x % (N_ / 16)) * 16;

  if (tid < 16) {
    int i = i0 + tid;
    const float* p = X + ((size_t)(b * N_ + i) * 5 + 1) * 3;  // CA atom
    float x = p[0], y = p[1], z = p[2];
    Q[tid*4+0] = x; Q[tid*4+1] = y; Q[tid*4+2] = z; Q[tid*4+3] = 0.f;
    qn[tid] = x*x + y*y + z*z;
    mq[tid] = mask[b * N_ + i];
  }
  __syncthreads();

  const int wid  = tid >> 5;
  const int lane = tid & 31;
  const int n    = lane & 15;
  const int kkb  = (lane < 16) ? 0 : 2;   // K-pair this lane holds (ISA 16x4 f32 A layout)

  v2f A;
  A.x = Q[n*4 + kkb];
  A.y = Q[n*4 + kkb + 1];   // K=3 slot is the zero pad

  for (int jt = wid; jt < N_ / 16; jt += 8) {
    int j = jt * 16 + n;
    const float* p = X + ((size_t)(b * N_ + j) * 5 + 1) * 3;
    float kx = p[0], ky = p[1], kz = p[2];
    float mj = mask[b * N_ + j];

    v2f Bv;                                 // B 4x16: row kkb / kkb+1, col n
    Bv.x = (kkb == 0) ? kx : kz;
    Bv.y = (kkb == 0) ? ky : 0.f;

    v8f C = {0.f,0.f,0.f,0.f,0.f,0.f,0.f,0.f};
    C = __builtin_amdgcn_wmma_f32_16x16x4_f32(false, A, false, Bv, (short)0, C,
                                              false, false);

    float kn = kx*kx + ky*ky + kz*kz;
    #pragma unroll
    for (int g = 0; g < 8; ++g) {
      int m = (lane < 16) ? g : g + 8;      // C/D layout: VGPR g -> rows g, g+8
      float d2 = fmaxf(qn[m] + kn - 2.f * C[g], 0.f);
      float d  = sqrtf(d2 + 1e-6f);
      float ok = mq[m] * mj;
      ldsD[m * N_ + jt * 16 + n] = (ok != 0.f) ? d : 1e30f;
    }
  }
  __syncthreads();

  // ---- top-K: each wave owns 2 rows ----
  for (int rr = 0; rr < 2; ++rr) {
    int r = wid * 2 + rr;
    int i = i0 + r;
    float* row = ldsD + r * N_;
    size_t base = (size_t)(b * N_ + i) * K_;
    for (int k = 0; k < K_; ++k) {
      unsigned long long best = ~0ull;       // distances >= 0 so float bits order == value order
      for (int j = lane; j < N_; j += 32) {
        unsigned long long key =
            ((unsigned long long)__float_as_uint(row[j]) << 32) | (unsigned)j;
        if (key < best) best = key;
      }
      #pragma unroll
      for (int off = 16; off > 0; off >>= 1) {
        unsigned long long o = __shfl_xor(best, off, 32);
        if (o < best) best = o;
      }
      int jm = (int)(best & 0xffffffffu);
      if (lane == 0) {
        dnbr[base + k]  = __uint_as_float((unsigned)(best >> 32));
        eidx[base + k]  = jm;
        eidxf[base + k] = (float)jm;
        row[jm] = __int_as_float(0x7f7fffff);   // FLT_MAX: remove from further rounds
      }
      asm volatile("s_wait_dscnt 0x0" ::: "memory");
    }
  }
}

// =====================================================================
// Kernel 2: per-node features. Block = 128 (4 waves), one wave / node.
// Scalar prologue (dihedrals, orientations) redundant across lanes;
// lanes split the 14x100 GEMV + 100-wide LayerNorm.
// =====================================================================
__global__ void k_node(const float* __restrict__ X,
                       const float* __restrict__ Wh,   // 3x8
                       const float* __restrict__ Ws,   // 14x100
                       const float* __restrict__ bs,   // 100
                       const float* __restrict__ Wv,   // 8x8
                       const float* __restrict__ gam, const float* __restrict__ bet,
                       float* __restrict__ Vout)
{
  __shared__ float lWs[14 * 100];
  __shared__ float lbs[100], lg[100], lb[100];
  const int tid = threadIdx.x;
  for (int idx = tid; idx < 1400; idx += 128) lWs[idx] = Ws[idx];
  for (int idx = tid; idx < 100;  idx += 128) { lbs[idx] = bs[idx]; lg[idx] = gam[idx]; lb[idx] = bet[idx]; }
  __syncthreads();

  const int wid = tid >> 5, lane = tid & 31;
  const int node = blockIdx.x * 4 + wid;
  const int b = node / N_, i = node % N_;

  auto atom = [&](int ii, int a) -> float3 {
    const float* p = X + ((size_t)(b * N_ + ii) * 5 + a) * 3;
    return f3(p[0], p[1], p[2]);
  };

  float3 ca  = atom(i, 1);
  float3 fwd = (i < N_ - 1) ? norm3(sub3(atom(i + 1, 1), ca)) : f3(0.f, 0.f, 0.f);
  float3 bwd = (i > 0)      ? scl3(norm3(sub3(ca, atom(i - 1, 1))), -1.f) : f3(0.f, 0.f, 0.f);

  float3 cN   = norm3(sub3(atom(i, 2), ca));
  float3 nN   = norm3(sub3(atom(i, 0), ca));
  float3 bis  = norm3(add3(cN, nN));
  float3 perp = norm3(cross3(cN, nN));
  float3 vec  = add3(scl3(bis, -0.57735026918962576f), scl3(perp, -0.81649658092772603f));

  // dihedrals: padded raw index t = 3i + d - 1, valid for 0 <= t <= 3N-4
  float dang[3];
  #pragma unroll
  for (int d = 0; d < 3; ++d) {
    int t = 3 * i + d - 1;
    float a = 0.f;
    if (t >= 0 && t <= 3 * N_ - 4) {
      float3 P0 = atom(t / 3, t % 3);
      float3 P1 = atom((t + 1) / 3, (t + 1) % 3);
      float3 P2 = atom((t + 2) / 3, (t + 2) % 3);
      float3 P3 = atom((t + 3) / 3, (t + 3) % 3);
      float3 u2 = norm3(sub3(P1, P0));
      float3 u1 = norm3(sub3(P2, P1));
      float3 u0 = norm3(sub3(P3, P2));
      float3 m2 = norm3(cross3(u2, u1));
      float3 m1 = norm3(cross3(u1, u0));
      float cd = fminf(fmaxf(dot3(m2, m1), -1.f + 1e-7f), 1.f - 1e-7f);
      float sg = dot3(u2, m1);
      sg = (sg > 0.f) ? 1.f : ((sg < 0.f) ? -1.f : 0.f);
      a = sg * acosf(cd);
    }
    dang[d] = a;
  }

  float sIn[14];
  sIn[0] = cosf(dang[0]); sIn[1] = cosf(dang[1]); sIn[2] = cosf(dang[2]);
  sIn[3] = sinf(dang[0]); sIn[4] = sinf(dang[1]); sIn[5] = sinf(dang[2]);

  float v[3][3] = { {vec.x, fwd.x, bwd.x},
                    {vec.y, fwd.y, bwd.y},
                    {vec.z, fwd.z, bwd.z} };
  float vh[3][8];
  #pragma unroll
  for (int r = 0; r < 3; ++r)
    #pragma unroll
    for (int h = 0; h < 8; ++h)
      vh[r][h] = v[r][0]*Wh[0*8+h] + v[r][1]*Wh[1*8+h] + v[r][2]*Wh[2*8+h];
  #pragma unroll
  for (int h = 0; h < 8; ++h)
    sIn[6 + h] = sqrtf(fmaxf(vh[0][h]*vh[0][h] + vh[1][h]*vh[1][h] + vh[2][h]*vh[2][h], 1e-8f));

  float vo[24];
  #pragma unroll
  for (int r = 0; r < 3; ++r)
    #pragma unroll
    for (int o = 0; o < 8; ++o) {
      float a = 0.f;
      #pragma unroll
      for (int h = 0; h < 8; ++h) a += vh[r][h] * Wv[h * 8 + o];
      vo[r * 8 + o] = a;
    }

  // lane-parallel 14x100 GEMV
  float acc[4];
  int cnt = 0;
  for (int c = lane; c < 100; c += 32) {
    float a = lbs[c];
    #pragma unroll
    for (int k = 0; k < 14; ++k) a += sIn[k] * lWs[k * 100 + c];
    acc[cnt++] = a;
  }
  float s = 0.f;
  for (int t = 0; t < cnt; ++t) s += acc[t];
  #pragma unroll
  for (int off = 16; off > 0; off >>= 1) s += __shfl_xor(s, off, 32);
  float mu = s * 0.01f;
  float vs = 0.f;
  for (int t = 0; t < cnt; ++t) { float dd = acc[t] - mu; vs += dd * dd; }
  #pragma unroll
  for (int off = 16; off > 0; off >>= 1) vs += __shfl_xor(vs, off, 32);
  float inv = 1.0f / sqrtf(vs * 0.01f + 1e-3f);

  size_t base = (size_t)node * 124;
  if (lane < 24) Vout[base + lane] = vo[lane];
  cnt = 0;
  for (int c = lane; c < 100; c += 32) {
    Vout[base + 24 + c] = (acc[cnt] - mu) * inv * lg[c] + lb[c];
    cnt++;
  }
}

// =====================================================================
// Kernel 3: per-edge features. Block = 256 (8 waves), one wave / edge.
// Lanes 0-15: RBF, 16-23: cos PE, 24-31: sin PE -> staged 33-vec in LDS;
// then lane c owns output channel c for the 33x32 GEMV and 32-wide LN.
// =====================================================================
__global__ void k_edge(const float* __restrict__ X,
                       const float* __restrict__ dnbr, const int* __restrict__ eidx,
                       const float* __restrict__ Whe, const float* __restrict__ Wse,  // (1,1),(33,32)
                       const float* __restrict__ bse, const float* __restrict__ Wve,  // (32),(1,1)
                       const float* __restrict__ ge,  const float* __restrict__ be,
                       float* __restrict__ Eout)
{
  __shared__ float lW[33 * 32];
  __shared__ float lbs[32], lg[32], lb[32];
  __shared__ float stage[8 * 33];
  const int tid = threadIdx.x;
  for (int idx = tid; idx < 33 * 32; idx += 256) lW[idx] = Wse[idx];
  if (tid < 32) { lbs[tid] = bse[tid]; lg[tid] = ge[tid]; lb[tid] = be[tid]; }
  __syncthreads();

  const int wid = tid >> 5, lane = tid & 31;
  const int e   = blockIdx.x * 8 + wid;
  const int b   = e / (N_ * K_);
  const int rem = e % (N_ * K_);
  const int i   = rem / K_;

  float d = dnbr[e];
  int   j = eidx[e];

  const float* pi = X + ((size_t)(b * N_ + i) * 5 + 1) * 3;
  const float* pj = X + ((size_t)(b * N_ + j) * 5 + 1) * 3;
  float dx = pj[0] - pi[0], dy = pj[1] - pi[1], dz = pj[2] - pi[2];
  float nn = sqrtf(fmaxf(dx*dx + dy*dy + dz*dz, 1e-8f));
  float ex = dx / nn, ey = dy / nn, ez = dz / nn;

  float w  = Whe[0];
  float vx = ex * w, vy = ey * w, vz = ez * w;
  float nrmE = sqrtf(fmaxf(vx*vx + vy*vy + vz*vz, 1e-8f));

  float sval;
  if (lane < 16) {
    float mu = (float)lane * (20.f / 15.f);       // linspace(0,20,16)
    float t  = (d - mu) * 0.8f;                   // / (20/16)
    sval = expf(-t * t);
  } else {
    int p   = (lane - 16) & 7;
    float f = expf((float)(2 * p) * -0.57564627324851148f);  // -ln(1e4)/16
    float ang = (float)(j - i) * f;
    sval = (lane < 24) ? cosf(ang) : sinf(ang);
  }
  stage[wid * 33 + lane] = sval;
  if (lane == 0) stage[wid * 33 + 32] = nrmE;
  asm volatile("s_wait_dscnt 0x0" ::: "memory");

  float acc = lbs[lane];
  #pragma unroll
  for (int k2 = 0; k2 < 33; ++k2) acc += stage[wid * 33 + k2] * lW[k2 * 32 + lane];

  float s = acc;
  #pragma unroll
  for (int off = 16; off > 0; off >>= 1) s += __shfl_xor(s, off, 32);
  float mu = s * (1.f / 32.f);
  float dd = acc - mu;
  float vs = dd * dd;
  #pragma unroll
  for (int off = 16; off > 0; off >>= 1) vs += __shfl_xor(vs, off, 32);
  float ln = dd * (1.0f / sqrtf(vs * (1.f / 32.f) + 1e-3f)) * lg[lane] + lb[lane];

  float wv = Wve[0];
  size_t base = (size_t)e * 35;
  if (lane == 0) { Eout[base] = vx * wv; Eout[base + 1] = vy * wv; Eout[base + 2] = vz * wv; }
  Eout[base + 3 + lane] = ln;
}

// =====================================================================
extern "C" void kernel_launch(void* const* d_in, const int* in_sizes, int n_in,
                              void* d_out, int out_size, void* d_ws, size_t ws_size,
                              hipStream_t stream) {
  const float* X    = (const float*)d_in[0];
  const float* mask = (const float*)d_in[1];
  const float* Wh_n = (const float*)d_in[2];
  const float* Ws_n = (const float*)d_in[3];
  const float* bs_n = (const float*)d_in[4];
  const float* Wv_n = (const float*)d_in[5];
  const float* Wh_e = (const float*)d_in[6];
  const float* Ws_e = (const float*)d_in[7];
  const float* bs_e = (const float*)d_in[8];
  const float* Wv_e = (const float*)d_in[9];
  const float* gn   = (const float*)d_in[10];
  const float* bn   = (const float*)d_in[11];
  const float* ge   = (const float*)d_in[12];
  const float* be   = (const float*)d_in[13];

  float* out = (float*)d_out;
  const size_t V_SZ = (size_t)B_ * N_ * 124;
  const size_t E_SZ = (size_t)B_ * N_ * K_ * 35;
  float* Vout  = out;
  float* Eout  = out + V_SZ;
  float* EidxF = out + V_SZ + E_SZ;

  float* dnbr = (float*)d_ws;
  int*   eidx = (int*)(dnbr + (size_t)B_ * N_ * K_);

  size_t shmem = (size_t)(16 * N_ + 96) * sizeof(float);   // 128KB stripe + staging
  k_dist_topk<<<B_ * (N_ / 16), 256, shmem, stream>>>(X, mask, dnbr, eidx, EidxF);
  k_node<<<(B_ * N_) / 4, 128, 0, stream>>>(X, Wh_n, Ws_n, bs_n, Wv_n, gn, bn, Vout);
  k_edge<<<(B_ * N_ * K_) / 8, 256, 0, stream>>>(X, dnbr, eidx, Wh_e, Ws_e, bs_e, Wv_e, ge, be, Eout);
}